// STDPLearner2_66511863546350
// MI455X (gfx1250) — compile-verified
//
#include <hip/hip_runtime.h>

typedef __attribute__((ext_vector_type(2))) float v2f;
typedef __attribute__((ext_vector_type(8))) float v8f;

#define T_STEPS 64
#define BATCH   16
#define PRE_N   4096
#define POST_N  4096
#define BT      128          // block tile edge (M = post, N = pre)
#define PAIRS   8            // BATCH / 2  (K stored pair-interleaved)
#define PR_STRIDE 288        // floats per pair-row: 2*128 + 32 pad
                             //   1152 B per pair-row; mod 256 == 128 so the two
                             //   lane-halves of a b64 frag load hit disjoint banks
#define TILE_F  (PAIRS * PR_STRIDE)

// LDS index of element (k, c) in pair-interleaved layout:
// one ds_load_b64 at (k>>1, c) returns {K=2p, K=2p+1} as an aligned v2f fragment.
__device__ __forceinline__ int pidx(int b, int c) {
  return (b >> 1) * PR_STRIDE + 2 * c + (b & 1);
}

// Fused STDP kernel:
//  - trace_pre / trace_post recurrences carried in LDS (one t == one K-chunk of 16)
//  - Delta_w accumulated with V_WMMA_F32_16X16X4_F32 (exact f32 path)
//  - double-buffered tiles -> single barrier per timestep
//  - S[post] = sum_{t,b} o*trace_post reduced in LDS, applied vs weight in epilogue
__global__ __launch_bounds__(256) void stdp_fused_kernel(
    const float* __restrict__ in_spikes,   // [T, B, PRE]
    const float* __restrict__ out_spikes,  // [T, B, POST]
    const float* __restrict__ weight,      // [POST, PRE]
    float* __restrict__ dw)                // [POST, PRE]
{
  __shared__ __align__(16) float Pl[2][TILE_F];  // trace_pre tiles  [buf][pair-interleaved]
  __shared__ __align__(16) float Ot[2][TILE_F];  // out_spikes tiles
  __shared__ __align__(16) float Tl[TILE_F];     // trace_post tile (owner-exclusive)
  __shared__ float Sred[256];
  __shared__ float Sl[BT];

  const int tid   = threadIdx.x;
  const int pre0  = blockIdx.x * BT;
  const int post0 = blockIdx.y * BT;

  // ---- recurrence ownership: each thread owns 8 (b, c) cells
  const int c     = tid & 127;
  const int bbase = (tid >> 7) * 8;

  #pragma unroll
  for (int j = 0; j < 8; ++j) {
    Pl[1][pidx(bbase + j, c)] = 0.0f;   // "previous" buffer for t = 0
    Tl[pidx(bbase + j, c)]    = 0.0f;
  }

  // ---- WMMA tiling: 8 wave32s as 2(M) x 4(N); wave tile = 64x32 = 4x2 tiles
  const int wid     = tid >> 5;          // wave32 on gfx1250
  const int waveM   = (wid >> 2) * 64;
  const int waveN   = (wid & 3) * 32;
  const int lanelow = tid & 15;
  const int half    = (tid >> 4) & 1;    // selects K-pair row (A/B) and M+8 (C/D)

  v8f acc[4][2];
  #pragma unroll
  for (int mt = 0; mt < 4; ++mt)
    #pragma unroll
    for (int nt = 0; nt < 2; ++nt)
      acc[mt][nt] = (v8f){0.f,0.f,0.f,0.f,0.f,0.f,0.f,0.f};

  float sacc = 0.0f;  // this thread's partial of S[post]

  const size_t inIdx  = (size_t)bbase * PRE_N  + pre0  + c;
  const size_t outIdx = (size_t)bbase * POST_N + post0 + c;

  for (int t = 0; t < T_STEPS; ++t) {
    const int buf = t & 1;

    // ---- stage timestep t into buffer `buf` (reads prev traces from buf^1;
    //      owner-exclusive cells, so no barrier needed before this)
    #pragma unroll
    for (int j = 0; j < 8; ++j) {
      const int    l  = pidx(bbase + j, c);
      const size_t gi = (size_t)t * (BATCH * PRE_N)  + inIdx  + (size_t)j * PRE_N;
      const size_t go = (size_t)t * (BATCH * POST_N) + outIdx + (size_t)j * POST_N;

      float iv = in_spikes[gi];
      float tp = 0.5f * Pl[buf ^ 1][l] + iv;        // trace - trace/2 + spike
      tp = fminf(fmaxf(tp, 0.0f), 1.0f);            // clip to [0,1]
      Pl[buf][l] = tp;

      float ov = out_spikes[go];
      Ot[buf][l] = ov;
      float tq = 0.5f * Tl[l] + ov;                 // trace_post (unclipped)
      Tl[l] = tq;
      sacc += ov * tq;                              // S[post] partial
    }

    if (t + 1 < T_STEPS) {                          // global_prefetch_b8 next step
      __builtin_prefetch(&in_spikes [(size_t)(t + 1) * (BATCH * PRE_N)  + inIdx ], 0, 0);
      __builtin_prefetch(&out_spikes[(size_t)(t + 1) * (BATCH * POST_N) + outIdx], 0, 0);
    }

    __syncthreads();   // publish buffer `buf`; also proves wmma(t-1) finished everywhere

    // ---- K = 16 (batch dim) as 4 WMMA k-steps of 4; fragments via ds_load_b64
    const v2f* OlP = (const v2f*)Ot[buf];   // float2 view: (pair, col) -> pair*144 + col
    const v2f* PlP = (const v2f*)Pl[buf];
    #pragma unroll
    for (int ks = 0; ks < 4; ++ks) {
      const int prow = (2 * ks + half) * (PR_STRIDE / 2);
      v2f a[4], b[2];
      #pragma unroll
      for (int mt = 0; mt < 4; ++mt)
        a[mt] = OlP[prow + waveM + mt * 16 + lanelow];   // A[m, 4ks+2h+{0,1}]
      #pragma unroll
      for (int nt = 0; nt < 2; ++nt)
        b[nt] = PlP[prow + waveN + nt * 16 + lanelow];   // B[4ks+2h+{0,1}, n]
      #pragma unroll
      for (int mt = 0; mt < 4; ++mt)
        #pragma unroll
        for (int nt = 0; nt < 2; ++nt)
          acc[mt][nt] = __builtin_amdgcn_wmma_f32_16x16x4_f32(
              false, a[mt], false, b[nt], (short)0, acc[mt][nt], false, false);
    }
  }

  // ---- finalize S[post_local]: reduce the two b-halves
  __syncthreads();
  Sred[tid] = sacc;
  __syncthreads();
  if (tid < BT) Sl[tid] = Sred[tid] + Sred[tid + BT];
  __syncthreads();

  // ---- epilogue: dw = acc - S[row] * weight   (C/D layout: vgpr v -> M = v + 8*half)
  #pragma unroll
  for (int mt = 0; mt < 4; ++mt) {
    #pragma unroll
    for (int v = 0; v < 8; ++v) {
      const int   lrow = waveM + mt * 16 + v + 8 * half;
      const float s    = Sl[lrow];
      const size_t rowb = (size_t)(post0 + lrow) * PRE_N;
      #pragma unroll
      for (int nt = 0; nt < 2; ++nt) {
        const size_t idx = rowb + pre0 + waveN + nt * 16 + lanelow;
        dw[idx] = acc[mt][nt][v] - s * weight[idx];
      }
    }
  }
}

extern "C" void kernel_launch(void* const* d_in, const int* in_sizes, int n_in,
                              void* d_out, int out_size, void* d_ws, size_t ws_size,
                              hipStream_t stream) {
  (void)in_sizes; (void)n_in; (void)out_size; (void)d_ws; (void)ws_size;
  const float* in_spikes  = (const float*)d_in[0];
  const float* out_spikes = (const float*)d_in[1];
  const float* weight     = (const float*)d_in[2];
  float* out = (float*)d_out;

  dim3 grid(PRE_N / BT, POST_N / BT);   // 32 x 32 blocks
  stdp_fused_kernel<<<grid, dim3(256), 0, stream>>>(in_spikes, out_spikes, weight, out);
}